// DCell_72584947302887
// MI455X (gfx1250) — compile-verified
//
#include <hip/hip_runtime.h>
#include <math.h>

typedef float v2f __attribute__((ext_vector_type(2)));
typedef float v8f __attribute__((ext_vector_type(8)));
typedef int   v2i __attribute__((ext_vector_type(2)));

#define EPSV 1e-5f

// Native gfx1250 V_TANH_F32 when available (single TRANS32 op).
#if __has_builtin(__builtin_amdgcn_tanhf)
__device__ __forceinline__ float fast_tanh(float x) { return __builtin_amdgcn_tanhf(x); }
#else
__device__ __forceinline__ float fast_tanh(float x) { return tanhf(x); }
#endif

// ---------------------------------------------------------------------------
// Zero-pad weights/bias to tile-aligned output count OUTP = NTILES*16 so the
// WMMA hot loop has NO per-lane predication (EXEC stays all-ones, loads fuse
// into global_load_b64 and clause cleanly).
//   Wp[s][n][k] = (n<OUT) ? W[s][n][k] : 0      n in [0,OUTP)
//   bp[s][n]    = (n<OUT) ? b[s][n]    : 0
// ---------------------------------------------------------------------------
__global__ void dcell_pad_wb(const float* __restrict__ W,
                             const float* __restrict__ bias,
                             float* __restrict__ Wp,
                             float* __restrict__ bp,
                             int S, int OUT, int OUTP, int KIN)
{
    const int i = blockIdx.x * blockDim.x + threadIdx.x;
    const int total = S * OUTP * KIN;
    if (i >= total) return;
    const int s = i / (OUTP * KIN);
    const int rem = i - s * OUTP * KIN;
    const int n = rem / KIN;
    const int k = rem - n * KIN;
    Wp[i] = (n < OUT) ? W[((size_t)s * OUT + n) * KIN + k] : 0.f;
    if (k == 0) bp[s * OUTP + n] = (n < OUT) ? bias[s * OUT + n] : 0.f;
}

// ---------------------------------------------------------------------------
// One "subsystem level": h = tanh(x @ W^T + b) -> hout, plus per-16-row-tile
// partial sums / sums-of-squares for the batch-norm statistics.
//
// Input operand x is formed on the fly:
//   OPREV == 0 : gather mode (level 1):  x[b][k] = mutant[b*6000 + gidx[s*16+k]]
//   OPREV  > 0 : concat of 8 children with folded BN affine:
//                c = k/OPREV, j = k%OPREV, ch = s*8+c
//                x[b][k] = scale[ch][j]*hprev[ch][b][j] + shift[ch][j]
//
// WMMA V_WMMA_F32_16X16X4_F32 fragments (wave32):
//   A (16x4): lane l -> m = l&15, k = k0 + 2*(l>>4) + {0,1}   (2 VGPRs)
//   B (4x16): lane l -> n = l&15, k = k0 + 2*(l>>4) + {0,1}   (B = W^T)
//   C/D     : vgpr r, lane l -> m = r + 8*(l>>4), n = l&15
//
// A (the large operand) is loaded ONCE per K-step and reused by all NTILES
// column tiles; weights come from the zero-padded Wp so the K loop is fully
// unpredicated.
// ---------------------------------------------------------------------------
template<int KIN, int OUT, int NTILES, int OPREV>
__global__ __launch_bounds__(128)
void dcell_level(const float* __restrict__ xsrc,   // mutant (OPREV==0) or hprev
                 const int*   __restrict__ gidx,   // [S,16]  (OPREV==0 only)
                 const float* __restrict__ scp,    // [S*8, OPREV] folded BN scale
                 const float* __restrict__ shp,    // [S*8, OPREV] folded BN shift
                 const float* __restrict__ Wp,     // [S, NTILES*16, KIN] padded
                 const float* __restrict__ bp,     // [S, NTILES*16]      padded
                 float* __restrict__ hout,         // [S, 4096, OUT]
                 float* __restrict__ psum,         // [S, 256, OUT]
                 float* __restrict__ psumsq)       // [S, 256, OUT]
{
    constexpr int OUTP = NTILES * 16;

    const int s    = blockIdx.y;
    const int wave = threadIdx.x >> 5;
    const int lane = threadIdx.x & 31;
    const int half = lane >> 4;
    const int lm   = lane & 15;
    const int rowBase = blockIdx.x * 64 + wave * 16;   // 64 batch rows / block
    const int b = rowBase + lm;                        // A-fragment row

    v8f acc[NTILES];
#pragma unroll
    for (int nt = 0; nt < NTILES; ++nt) {
        const float bval = bp[(size_t)s * OUTP + nt * 16 + lm];
#pragma unroll
        for (int r = 0; r < 8; ++r) acc[nt][r] = bval;  // C = broadcast bias[n]
    }

    // per-lane base pointer into this subsystem's padded weights (row n=lm)
    const float* wrow = Wp + ((size_t)s * OUTP + lm) * KIN;

    for (int k0 = 0; k0 < KIN; k0 += 4) {
        const int k = k0 + half * 2;
        v2f a;
        if constexpr (OPREV == 0) {
            const v2i g = *(const v2i*)(gidx + s * 16 + k);   // 8B-aligned pair
            a.x = xsrc[(size_t)b * 6000 + g.x];
            a.y = xsrc[(size_t)b * 6000 + g.y];
        } else {
            const int c0 = k / OPREV,       j0 = k - c0 * OPREV;
            const int c1 = (k + 1) / OPREV, j1 = (k + 1) - c1 * OPREV;
            const int ch0 = s * 8 + c0, ch1 = s * 8 + c1;
            const float x0 = xsrc[((size_t)ch0 * 4096 + b) * OPREV + j0];
            const float x1 = xsrc[((size_t)ch1 * 4096 + b) * OPREV + j1];
            a.x = fmaf(scp[ch0 * OPREV + j0], x0, shp[ch0 * OPREV + j0]);
            a.y = fmaf(scp[ch1 * OPREV + j1], x1, shp[ch1 * OPREV + j1]);
        }
#pragma unroll
        for (int nt = 0; nt < NTILES; ++nt) {
            const v2f bb = *(const v2f*)(wrow + (size_t)nt * 16 * KIN + k); // b64
            acc[nt] = __builtin_amdgcn_wmma_f32_16x16x4_f32(
                false, a, false, bb, (short)0, acc[nt], false, false);
        }
    }

    // tanh, store h, and reduce per-column sums for batch-norm stats.
#pragma unroll
    for (int nt = 0; nt < NTILES; ++nt) {
        const int  n  = nt * 16 + lm;
        const bool nv = (n < OUT);                     // folds to true when aligned
        float sum = 0.f, sq = 0.f;
#pragma unroll
        for (int r = 0; r < 8; ++r) {
            const float h = fast_tanh(acc[nt][r]);
            const int   m = rowBase + r + half * 8;
            if (nv) hout[((size_t)s * 4096 + m) * OUT + n] = h;
            sum += h; sq += h * h;
        }
        // combine the two half-rows (lane and lane^16 hold the same column n)
        sum += __shfl_xor(sum, 16, 32);
        sq  += __shfl_xor(sq,  16, 32);
        if (half == 0 && nv) {
            const int t16 = blockIdx.x * 4 + wave;     // 16-row tile id, 0..255
            psum  [((size_t)s * 256 + t16) * OUT + n] = sum;
            psumsq[((size_t)s * 256 + t16) * OUT + n] = sq;
        }
    }
}

// Deterministic (ordered) reduction of the 256 per-tile partials; folds the
// batch-norm into   scale = g * rsqrt(var+eps),  shift = beta - mean*scale.
__global__ void dcell_stats(const float* __restrict__ psum,
                            const float* __restrict__ psumsq,
                            const float* __restrict__ g,
                            const float* __restrict__ beta,
                            float* __restrict__ scale,
                            float* __restrict__ shift,
                            int total, int OUT)
{
    const int i = blockIdx.x * blockDim.x + threadIdx.x;
    if (i >= total) return;
    const int s = i / OUT, n = i - s * OUT;
    float sum = 0.f, sq = 0.f;
    for (int t = 0; t < 256; ++t) {
        const size_t base = ((size_t)s * 256 + t) * OUT + n;
        sum += psum[base];
        sq  += psumsq[base];
    }
    const float mean = sum * (1.f / 4096.f);
    const float var  = sq  * (1.f / 4096.f) - mean * mean;
    const float sc   = g[i] * rsqrtf(var + EPSV);
    scale[i] = sc;
    shift[i] = beta[i] - mean * sc;
}

// Final affine: out = hr * scale + shift   ([4096,64])
__global__ void dcell_final(const float* __restrict__ hr,
                            const float* __restrict__ scale,
                            const float* __restrict__ shift,
                            float* __restrict__ out)
{
    const int i = blockIdx.x * 256 + threadIdx.x;
    const int n = i & 63;
    out[i] = fmaf(hr[i], scale[n], shift[n]);
}

extern "C" void kernel_launch(void* const* d_in, const int* in_sizes, int n_in,
                              void* d_out, int out_size, void* d_ws, size_t ws_size,
                              hipStream_t stream)
{
    const float* mutant = (const float*)d_in[0];
    const int*   gidx   = (const int*)  d_in[1];
    const float* W1 = (const float*)d_in[2];  const float* b1  = (const float*)d_in[3];
    const float* g1 = (const float*)d_in[4];  const float* be1 = (const float*)d_in[5];
    const float* W2 = (const float*)d_in[6];  const float* b2  = (const float*)d_in[7];
    const float* g2 = (const float*)d_in[8];  const float* be2 = (const float*)d_in[9];
    const float* W3 = (const float*)d_in[10]; const float* b3  = (const float*)d_in[11];
    const float* g3 = (const float*)d_in[12]; const float* be3 = (const float*)d_in[13];
    const float* Wr = (const float*)d_in[14]; const float* br  = (const float*)d_in[15];
    const float* gr = (const float*)d_in[16]; const float* ber = (const float*)d_in[17];
    float* out = (float*)d_out;

    // workspace carve (floats)
    float* p = (float*)d_ws;
    auto carve = [&](size_t nf) { float* r = p; p += nf; return r; };
    float* h1  = carve(512ull * 4096 * 20);
    float* h2  = carve(64ull  * 4096 * 24);
    float* h3  = carve(8ull   * 4096 * 32);
    float* hr  = carve(4096ull * 64);
    float* ps1 = carve(512ull * 256 * 20); float* pq1 = carve(512ull * 256 * 20);
    float* ps2 = carve(64ull  * 256 * 24); float* pq2 = carve(64ull  * 256 * 24);
    float* ps3 = carve(8ull   * 256 * 32); float* pq3 = carve(8ull   * 256 * 32);
    float* psr = carve(256ull * 64);       float* pqr = carve(256ull * 64);
    float* sc1 = carve(512 * 20); float* sh1 = carve(512 * 20);
    float* sc2 = carve(64 * 24);  float* sh2 = carve(64 * 24);
    float* sc3 = carve(8 * 32);   float* sh3 = carve(8 * 32);
    float* scr = carve(64);       float* shr = carve(64);
    float* Wp1 = carve(512ull * 32 * 16);  float* bp1 = carve(512 * 32);
    float* Wp2 = carve(64ull  * 32 * 160); float* bp2 = carve(64 * 32);

    // Pad L1/L2 weights+bias to 16-aligned output tiles (L3: 32, root: 64 are
    // already aligned and are used directly).
    dcell_pad_wb<<<(512 * 32 * 16 + 255) / 256, 256, 0, stream>>>(
        W1, b1, Wp1, bp1, 512, 20, 32, 16);
    dcell_pad_wb<<<(64 * 32 * 160 + 255) / 256, 256, 0, stream>>>(
        W2, b2, Wp2, bp2, 64, 24, 32, 160);

    // Level 1: gather 16 genes, GEMM vs W1[512,20,16]
    dcell_level<16, 20, 2, 0><<<dim3(64, 512), 128, 0, stream>>>(
        mutant, gidx, nullptr, nullptr, Wp1, bp1, h1, ps1, pq1);
    dcell_stats<<<(512 * 20 + 255) / 256, 256, 0, stream>>>(
        ps1, pq1, g1, be1, sc1, sh1, 512 * 20, 20);

    // Level 2: 8x20 children concat -> 160, GEMM vs W2[64,24,160]
    dcell_level<160, 24, 2, 20><<<dim3(64, 64), 128, 0, stream>>>(
        h1, nullptr, sc1, sh1, Wp2, bp2, h2, ps2, pq2);
    dcell_stats<<<(64 * 24 + 255) / 256, 256, 0, stream>>>(
        ps2, pq2, g2, be2, sc2, sh2, 64 * 24, 24);

    // Level 3: 8x24 -> 192, GEMM vs W3[8,32,192]  (OUT=32 already aligned)
    dcell_level<192, 32, 2, 24><<<dim3(64, 8), 128, 0, stream>>>(
        h2, nullptr, sc2, sh2, W3, b3, h3, ps3, pq3);
    dcell_stats<<<(8 * 32 + 255) / 256, 256, 0, stream>>>(
        ps3, pq3, g3, be3, sc3, sh3, 8 * 32, 32);

    // Root: 8x32 -> 256, GEMM vs Wr[64,256]       (OUT=64 already aligned)
    dcell_level<256, 64, 4, 32><<<dim3(64, 1), 128, 0, stream>>>(
        h3, nullptr, sc3, sh3, Wr, br, hr, psr, pqr);
    dcell_stats<<<1, 64, 0, stream>>>(psr, pqr, gr, ber, scr, shr, 64, 64);

    // out = gr*(hr-mu)*rsqrt(var+eps)+betar, folded as affine
    dcell_final<<<(4096 * 64) / 256, 256, 0, stream>>>(hr, scr, shr, out);
}